// RelGCN_32229434589747
// MI455X (gfx1250) — compile-verified
//
#include <hip/hip_runtime.h>
#include <hip/hip_bf16.h>

typedef __attribute__((ext_vector_type(2))) float v2f;
typedef __attribute__((ext_vector_type(8))) float v8f;

#define HFEAT 64   // HID_F == OUT_F == 64 in the reference

// ---------------------------------------------------------------------------
// small elementwise helpers
// ---------------------------------------------------------------------------
__global__ void zero_u32(unsigned* __restrict__ p, int n) {
  int i = blockIdx.x * blockDim.x + threadIdx.x;
  if (i < n) p[i] = 0u;
}

__global__ void relu_kernel(float* __restrict__ p, int n) {
  int i = blockIdx.x * blockDim.x + threadIdx.x;
  if (i < n) p[i] = fmaxf(p[i], 0.0f);
}

// count edges per (dst, relation) key -> u32 atomics, cnt fits in L2 (1.6 MB)
__global__ void count_edges(const int* __restrict__ dst,
                            const int* __restrict__ et,
                            unsigned* __restrict__ cnt, int E, int R) {
  int e = blockIdx.x * blockDim.x + threadIdx.x;
  if (e < E) atomicAdd(&cnt[dst[e] * R + et[e]], 1u);
}

// ---------------------------------------------------------------------------
// fp32 WMMA GEMM: Out[r] = A[N x K] @ Wall[r][K x 64] (+ bias)
// One wave computes a 16x64 output strip: 4 accumulators reuse one A fragment.
// V_WMMA_F32_16X16X4_F32 fragment layouts (ISA 7.12.2):
//   A 16x4 : lane L holds row (L&15), K = (L>>4)*2 + {vgpr0,vgpr1}
//   B 4x16 : lane L holds col (L&15), K = (L>>4)*2 + {vgpr0,vgpr1}
//   C 16x16: lane L, vgpr v -> row v + (L>>4)*8, col (L&15)
// ---------------------------------------------------------------------------
__global__ __launch_bounds__(128) void wmma_gemm_strip(
    const float* __restrict__ A,     // [N, K] row-major
    const float* __restrict__ Wall,  // [gridDim.y, K, 64]
    const float* __restrict__ bias,  // [64] or nullptr
    float* __restrict__ Out,         // [gridDim.y, N, 64]
    int Nn, int K) {
  const int H = HFEAT;
  int wave = threadIdx.x >> 5;
  int lane = threadIdx.x & 31;
  int n_tiles = (Nn + 15) >> 4;
  int tn = blockIdx.x * (blockDim.x >> 5) + wave;
  if (tn >= n_tiles) return;  // whole wave exits together (EXEC stays all-1s)

  int r = blockIdx.y;
  const float* W = Wall + (size_t)r * K * H;
  float* O = Out + (size_t)r * Nn * H;

  int mrow = tn * 16 + (lane & 15);
  if (mrow >= Nn) mrow = Nn - 1;   // clamp for ragged tail (N=50000 is exact)
  int ksel = (lane >> 4) << 1;     // 0 for lanes 0-15, 2 for lanes 16-31
  int hcol = lane & 15;

  v8f c0 = {}, c1 = {}, c2 = {}, c3 = {};
  for (int k = 0; k < K; k += 4) {
    const float* arow = A + (size_t)mrow * K + k + ksel;
    v2f a;
    a.x = arow[0];
    a.y = arow[1];
    const float* w0 = W + (size_t)(k + ksel) * H;  // K row (k+ksel)
    const float* w1 = w0 + H;                      // K row (k+ksel+1)
    v2f b0, b1, b2, b3;
    b0.x = w0[hcol];      b0.y = w1[hcol];
    b1.x = w0[hcol + 16]; b1.y = w1[hcol + 16];
    b2.x = w0[hcol + 32]; b2.y = w1[hcol + 32];
    b3.x = w0[hcol + 48]; b3.y = w1[hcol + 48];
    c0 = __builtin_amdgcn_wmma_f32_16x16x4_f32(false, a, false, b0, (short)0, c0, false, false);
    c1 = __builtin_amdgcn_wmma_f32_16x16x4_f32(false, a, false, b1, (short)0, c1, false, false);
    c2 = __builtin_amdgcn_wmma_f32_16x16x4_f32(false, a, false, b2, (short)0, c2, false, false);
    c3 = __builtin_amdgcn_wmma_f32_16x16x4_f32(false, a, false, b3, (short)0, c3, false, false);
  }

  v8f acc[4] = {c0, c1, c2, c3};
  int rbase = tn * 16 + ((lane >> 4) << 3);
#pragma unroll
  for (int t = 0; t < 4; ++t) {
    int col = t * 16 + hcol;
    float bv = bias ? bias[col] : 0.0f;
#pragma unroll
    for (int v = 0; v < 8; ++v) {
      int row = rbase + v;
      if (row < Nn) O[(size_t)row * H + col] = acc[t][v] + bv;
    }
  }
}

// ---------------------------------------------------------------------------
// edge scatter: one wave per edge; lane f loads float2 of xw[etype][src],
// scales by 1/cnt, hardware-fp32-atomic-adds into out[dst] (L2-resident).
// ---------------------------------------------------------------------------
__global__ void scatter_edges(const int* __restrict__ src,
                              const int* __restrict__ dst,
                              const int* __restrict__ et,
                              const unsigned* __restrict__ cnt,
                              const float* __restrict__ xw,  // [R, N, 64]
                              float* __restrict__ out,       // [N, 64]
                              int E, int Nn, int R) {
  int gid = blockIdx.x * blockDim.x + threadIdx.x;
  int e = gid >> 5;
  if (e >= E) return;
  int lane = gid & 31;
  int s = src[e];
  int d = dst[e];
  int r = et[e];
  unsigned c = cnt[d * R + r];
  float w = 1.0f / (float)(c ? c : 1u);
  const float2* p = (const float2*)(xw + ((size_t)r * Nn + s) * HFEAT);
  float2 v = p[lane];
  float* o = out + (size_t)d * HFEAT + lane * 2;
  unsafeAtomicAdd(o, v.x * w);
  unsafeAtomicAdd(o + 1, v.y * w);
}

// ---------------------------------------------------------------------------
// launch
// ---------------------------------------------------------------------------
extern "C" void kernel_launch(void* const* d_in, const int* in_sizes, int n_in,
                              void* d_out, int out_size, void* d_ws, size_t ws_size,
                              hipStream_t stream) {
  const float* x       = (const float*)d_in[0];
  const int*   eidx    = (const int*)d_in[1];   // [2, E]: row 0 = src, row 1 = dst
  const int*   etype   = (const int*)d_in[2];   // [E]
  const float* W1_rel  = (const float*)d_in[3]; // [R, 128, 64]
  const float* W1_root = (const float*)d_in[4]; // [128, 64]
  const float* b1      = (const float*)d_in[5]; // [64]
  const float* W2_rel  = (const float*)d_in[6]; // [R, 64, 64]
  const float* W2_root = (const float*)d_in[7]; // [64, 64]
  const float* b2      = (const float*)d_in[8]; // [64]
  float* out = (float*)d_out;

  const int IN_F = 128;
  const int Nn = in_sizes[0] / IN_F;            // 50000
  const int E  = in_sizes[2];                   // 1600000
  const int R  = in_sizes[3] / (IN_F * HFEAT);  // 8

  const int* srcp = eidx;
  const int* dstp = eidx + E;

  // workspace layout: cnt (N*R u32) | xw (R*N*64 f32, ~102 MB) | h (N*64 f32)
  char* ws = (char*)d_ws;
  unsigned* cnt = (unsigned*)ws;
  size_t off = (((size_t)Nn * R * sizeof(unsigned)) + 255) & ~(size_t)255;
  float* xw = (float*)(ws + off);
  off += (size_t)R * Nn * HFEAT * sizeof(float);
  float* h = (float*)(ws + off);

  int n_tiles = (Nn + 15) / 16;              // 3125
  int gemm_blocks = (n_tiles + 3) / 4;       // 4 waves (128 thr) per block
  long long sc_threads = (long long)E * 32;  // one wave per edge
  unsigned sc_blocks = (unsigned)((sc_threads + 255) / 256);

  // --- degree counts (shared by both layers) ---
  zero_u32<<<(Nn * R + 255) / 256, 256, 0, stream>>>(cnt, Nn * R);
  count_edges<<<(E + 255) / 256, 256, 0, stream>>>(dstp, etype, cnt, E, R);

  // --- layer 1 ---
  wmma_gemm_strip<<<dim3(gemm_blocks, 1), 128, 0, stream>>>(x, W1_root, b1, h, Nn, IN_F);
  wmma_gemm_strip<<<dim3(gemm_blocks, R), 128, 0, stream>>>(x, W1_rel, nullptr, xw, Nn, IN_F);
  scatter_edges<<<sc_blocks, 256, 0, stream>>>(srcp, dstp, etype, cnt, xw, h, E, Nn, R);
  relu_kernel<<<(Nn * HFEAT + 255) / 256, 256, 0, stream>>>(h, Nn * HFEAT);

  // --- layer 2 ---
  wmma_gemm_strip<<<dim3(gemm_blocks, 1), 128, 0, stream>>>(h, W2_root, b2, out, Nn, HFEAT);
  wmma_gemm_strip<<<dim3(gemm_blocks, R), 128, 0, stream>>>(h, W2_rel, nullptr, xw, Nn, HFEAT);
  scatter_edges<<<sc_blocks, 256, 0, stream>>>(srcp, dstp, etype, cnt, xw, out, E, Nn, R);
}